// NormDistConv_3908420239443
// MI455X (gfx1250) — compile-verified
//
#include <hip/hip_runtime.h>
#include <stdint.h>

#define IN_C   64
#define OUT_C  64
#define HW     32
#define KDIM   576            // 64*3*3, im2col ordering (c,kh,kw)
#define XS_ROWSTRIDE 34       // 32 cols + 2 pad cols
#define XS_CSTRIDE  (3 * XS_ROWSTRIDE)

typedef unsigned int v4u __attribute__((ext_vector_type(4)));
typedef int          v8i __attribute__((ext_vector_type(8)));
typedef int          v4i __attribute__((ext_vector_type(4)));

// ---------------------------------------------------------------------------
// Pass 1: transpose weights [O][Kdim] -> wT [Kdim][O] so the main kernel's
// per-block weight tile (16 consecutive channels, all k) is a contiguous-row
// strided 2D tile the TDM can fetch directly.
// ---------------------------------------------------------------------------
__global__ __launch_bounds__(256) void wt_transpose_kernel(
    const float* __restrict__ w, float* __restrict__ wT) {
  int i = blockIdx.x * 256 + threadIdx.x;       // i < KDIM*OUT_C = 36864
  int o = i & 63;
  int k = i >> 6;
  wT[i] = w[o * KDIM + k];                      // coalesced writes
}

// ---------------------------------------------------------------------------
// Main kernel. Block = (channel-group cg of 16, output row y, batch b).
// 128 threads = 4 waves; thread t: col = t&31, og = t>>5 (4 channels each).
// Weights arrive via Tensor Data Mover (async DMA -> LDS); x slab is staged
// cooperatively with zero-padded borders.
// ---------------------------------------------------------------------------
__global__ __launch_bounds__(128) void normdistconv_kernel(
    const float* __restrict__ x, const float* __restrict__ wT,
    const float* __restrict__ bias, float* __restrict__ out) {
  __shared__ __align__(16) float ws[KDIM * 16];            // [k][16]  36,864 B
  __shared__ __align__(16) float xs[IN_C * XS_CSTRIDE];    // [c][3][34] 26,112 B

  const int tid = threadIdx.x;
  const int cg  = blockIdx.x;        // 0..3  -> channels cg*16 .. cg*16+15
  const int y   = blockIdx.y;        // 0..31 output row
  const int b   = blockIdx.z;        // 0..1

  // ---- TDM: DMA the 576x16 weight tile (stride-64 rows) into LDS ----------
#if __has_builtin(__builtin_amdgcn_tensor_load_to_lds)
  if (tid < 32) {                    // one wave issues the descriptor
    uint64_t ga = (uint64_t)(uintptr_t)(wT + cg * 16);
    uint32_t lds_ws = (uint32_t)(uintptr_t)(void*)&ws[0];  // LDS aperture low bits
    v4u g0 = { 1u,                                  // count=1, user descriptor
               lds_ws,                              // lds_addr (bytes)
               (uint32_t)ga,                        // global_addr[31:0]
               (uint32_t)((ga >> 32) & 0x01FFFFFFu) | (2u << 30) }; // addr[56:32] | type=2
    v8i g1 = { (int)(2u << 16),      // data_size=4B; no mask/pad/iterate
               (int)(64u << 16),     // tensor_dim0 = 64 (bits 79:48, low half)
               (int)(576u << 16),    // dim0 hi=0 | tensor_dim1 = 576 (bits 111:80 low)
               (int)(16u << 16),     // dim1 hi=0 | tile_dim0 = 16 (bits 127:112)
               (int)576,             // tile_dim1 = 576, tile_dim2 = 0
               (int)64,              // tensor_dim0_stride = 64 elements (48b, low)
               0, 0 };               // stride0 hi | tensor_dim1_stride = 0
    v4i gz4 = { 0, 0, 0, 0 };
    v8i gz8 = { 0, 0, 0, 0, 0, 0, 0, 0 };
    __builtin_amdgcn_tensor_load_to_lds(g0, g1, gz4, gz4, gz8, 0);
  }
#endif

  // ---- cooperative staging of padded x slab (rows y-1..y+1, 64 ch) --------
  const float* xb = x + (size_t)b * IN_C * HW * HW;
  for (int i = tid; i < IN_C * 3 * HW; i += 128) {
    int c  = i / (3 * HW);
    int r  = (i / HW) % 3;
    int cc = i & (HW - 1);
    int srow = y - 1 + r;
    float v = ((unsigned)srow < (unsigned)HW) ? xb[(c * HW + srow) * HW + cc] : 0.f;
    xs[c * XS_CSTRIDE + r * XS_ROWSTRIDE + 1 + cc] = v;
  }
  for (int i = tid; i < IN_C * 3; i += 128) {              // left/right pad cols
    int c = i / 3, r = i % 3;
    xs[c * XS_CSTRIDE + r * XS_ROWSTRIDE + 0]  = 0.f;
    xs[c * XS_CSTRIDE + r * XS_ROWSTRIDE + 33] = 0.f;
  }

#if __has_builtin(__builtin_amdgcn_tensor_load_to_lds)
  if (tid < 32) {
  #if __has_builtin(__builtin_amdgcn_s_wait_tensorcnt)
    __builtin_amdgcn_s_wait_tensorcnt(0);                  // TDM done -> LDS valid
  #endif
  }
#else
  // fallback path if TDM builtin is unavailable on this toolchain
  for (int i = tid; i < KDIM * 16; i += 128) {
    int k = i >> 4, j = i & 15;
    ws[i] = wT[k * 64 + cg * 16 + j];
  }
#endif
  __syncthreads();

  // ---- compute: 4 accumulators per thread, Σ (x-w)^8 over 576 taps --------
  const int col = tid & 31;
  const int og  = tid >> 5;                                // wave-uniform
  const float4* wp = (const float4*)ws;                    // ws[k*16 + og*4]
  float a0 = 0.f, a1 = 0.f, a2 = 0.f, a3 = 0.f;

  for (int c = 0; c < IN_C; ++c) {
    const float* xr = &xs[c * XS_CSTRIDE + col];
#pragma unroll
    for (int kh = 0; kh < 3; ++kh) {
      float x0 = xr[kh * XS_ROWSTRIDE + 0];
      float x1 = xr[kh * XS_ROWSTRIDE + 1];
      float x2 = xr[kh * XS_ROWSTRIDE + 2];
      int kbase = c * 9 + kh * 3;
#pragma unroll
      for (int kw = 0; kw < 3; ++kw) {
        float xv = (kw == 0) ? x0 : ((kw == 1) ? x1 : x2);
        float4 w = wp[(size_t)(kbase + kw) * 4 + og];      // ds_load_b128 broadcast
        float d;
        d = xv - w.x; d = d * d; d = d * d; a0 = fmaf(d, d, a0);
        d = xv - w.y; d = d * d; d = d * d; a1 = fmaf(d, d, a1);
        d = xv - w.z; d = d * d; d = d * d; a2 = fmaf(d, d, a2);
        d = xv - w.w; d = d * d; d = d * d; a3 = fmaf(d, d, a3);
      }
    }
  }

  // y = (Σ d^8)^(1/8) + bias  (== reference's L∞-normalized form exactly)
  const int obase = cg * 16 + og * 4;
  float r0 = sqrtf(sqrtf(sqrtf(a0))) + bias[obase + 0];
  float r1 = sqrtf(sqrtf(sqrtf(a1))) + bias[obase + 1];
  float r2 = sqrtf(sqrtf(sqrtf(a2))) + bias[obase + 2];
  float r3 = sqrtf(sqrtf(sqrtf(a3))) + bias[obase + 3];

  float* ob = out + (((size_t)b * OUT_C + obase) * HW + y) * HW + col;
  ob[0 * HW * HW] = r0;
  ob[1 * HW * HW] = r1;
  ob[2 * HW * HW] = r2;
  ob[3 * HW * HW] = r3;
}

// ---------------------------------------------------------------------------
extern "C" void kernel_launch(void* const* d_in, const int* in_sizes, int n_in,
                              void* d_out, int out_size, void* d_ws, size_t ws_size,
                              hipStream_t stream) {
  const float* x    = (const float*)d_in[0];  // [2,64,32,32]
  const float* w    = (const float*)d_in[1];  // [64,64,3,3]
  const float* bias = (const float*)d_in[2];  // [64]
  float* out = (float*)d_out;                 // [2,64,32,32]
  float* wT  = (float*)d_ws;                  // scratch: 36864 floats

  wt_transpose_kernel<<<dim3((KDIM * OUT_C) / 256), dim3(256), 0, stream>>>(w, wT);
  normdistconv_kernel<<<dim3(4, HW, 2), dim3(128), 0, stream>>>(x, wT, bias, out);
}